// LIF_51436528337048
// MI455X (gfx1250) — compile-verified
//
#include <hip/hip_runtime.h>
#include <cstdint>

// ---------------------------------------------------------------------------
// LIF scan over T with TDM (tensor_load_to_lds) double-buffered staging.
// z: [B=32, T=1024, H=512] f32  ->  out: same shape, out[:,0,:]=0,
// out[:,t,:] = heaviside(V_t - 1),  V_t = 0.9 V_{t-1} + z[:,t-1,:] - (V_{t-1}>1)
// Memory-bound streaming scan: 128 MiB @ 23.3 TB/s ~= 5.6 us floor.
// Single-wave workgroups: each wave drives its own TDM pipeline, no barriers.
// ---------------------------------------------------------------------------

typedef unsigned int u32x4 __attribute__((ext_vector_type(4)));
typedef int          i32x8 __attribute__((ext_vector_type(8)));
typedef int          i32x4 __attribute__((ext_vector_type(4)));
typedef float        f32x2 __attribute__((ext_vector_type(2)));

constexpr int kB     = 32;
constexpr int kT     = 1024;
constexpr int kH     = 512;
constexpr int VEC    = 2;             // columns per lane (b64 LDS/global traffic)
constexpr int NT     = 32;            // one wave per block
constexpr int HT     = NT * VEC;      // 64 h-columns per block
constexpr int CHUNK  = 32;            // timesteps per DMA tile (8 KB)
constexpr int NCHUNK = kT / CHUNK;    // 32 chunks cover input rows 0..1023

// Issue one 2D TDM tile load: CHUNK rows x HT floats, row stride kH floats,
// from global address `gaddr` into LDS byte offset `lds_off`.
// D# layout per CDNA5 ISA 8.3-8.5 (group0: count/lds/global/type,
// group1: data_size/dims/strides, groups 2-3 unused for a 2D tile).
__device__ __forceinline__ void tdm_load_tile(unsigned lds_off,
                                              unsigned long long gaddr) {
  u32x4 g0;
  g0.x = 1u;                                              // count=1 (valid), user mode
  g0.y = lds_off;                                         // lds_addr (bytes)
  g0.z = (unsigned)(gaddr & 0xFFFFFFFFull);               // global_addr[31:0]
  g0.w = (unsigned)((gaddr >> 32) & 0x01FFFFFFull)        // global_addr[56:32]
       | (2u << 30);                                      // type = 2 ("image")
  i32x8 g1;
  g1[0] = (int)(2u << 16);                // data_size=2 (4B); wg_mask=0; no pad/iter
  g1[1] = (int)((unsigned)HT << 16);      // tensor_dim0[15:0] = HT (tile in-bounds)
  g1[2] = (int)((unsigned)CHUNK << 16);   // tensor_dim1[15:0] = CHUNK
  g1[3] = (int)((unsigned)HT << 16);      // tile_dim0 = HT
  g1[4] = (int)(unsigned)CHUNK;           // tile_dim1 = CHUNK; tile_dim2 = 0 (2D)
  g1[5] = (int)(unsigned)kH;              // tensor_dim0_stride[31:0] = 512 elems
  g1[6] = 0;                              // stride0[47:32]=0; stride1[15:0]=0
  g1[7] = 0;                              // stride1[47:16]=0
  i32x4 gz4 = {0, 0, 0, 0};               // groups 2/3 unused for 2D tile
  i32x8 gz8 = {0, 0, 0, 0, 0, 0, 0, 0};   // trailing group (6-arg toolchain form)
  __builtin_amdgcn_tensor_load_to_lds(g0, g1, gz4, gz4, gz8, 0);
}

__global__ __launch_bounds__(NT) void lif_tdm_kernel(const float* __restrict__ z,
                                                     float* __restrict__ out) {
  __shared__ float sbuf[2][CHUNK * HT];   // 2 x 8 KB double buffer

  const int b   = blockIdx.x;             // 0..31
  const int hb  = blockIdx.y * HT;        // h tile base
  const int tid = threadIdx.x;            // 0..31; owns columns hb + 2*tid + {0,1}

  const float* zbase = z   + ((size_t)b * kT) * kH + hb;
  float*       obase = out + ((size_t)b * kT) * kH + hb;

  // Defined-initialize sbuf so the (TDM-written, alias-invisible) reads can't
  // be folded away; b64 stores keep it cheap.
  const f32x2 zero2 = {0.0f, 0.0f};
  #pragma unroll
  for (int i = 0; i < 2 * CHUNK; ++i)
    *(f32x2*)&sbuf[0][i * HT + VEC * tid] = zero2;

  // t = 0 output row is zero by construction.
  __builtin_nontemporal_store(zero2, (f32x2*)&obase[VEC * tid]);

  // Init ds_stores must land before the DMA engine writes the same LDS.
  asm volatile("s_wait_dscnt 0x0" ::: "memory");

  // Prologue: DMA chunk 0 (input rows 0..31) into buffer 0.
  tdm_load_tile((unsigned)(uintptr_t)&sbuf[0][0],
                (unsigned long long)(uintptr_t)zbase);

  f32x2 V = {0.0f, 0.0f};
  for (int c = 0; c < NCHUNK; ++c) {
    const int nb = c + 1;
    if (nb < NCHUNK) {
      // All prior LDS reads done -> buf[nb&1] is free for the DMA engine.
      asm volatile("s_wait_dscnt 0x0" ::: "memory");
      tdm_load_tile((unsigned)(uintptr_t)&sbuf[nb & 1][0],
                    (unsigned long long)(uintptr_t)
                        (zbase + (size_t)nb * CHUNK * kH));
      // TDM ops are in-order per wave: <=1 outstanding => chunk c landed.
      __builtin_amdgcn_s_wait_tensorcnt((short)1);
    } else {
      __builtin_amdgcn_s_wait_tensorcnt((short)0);
    }
    asm volatile("" ::: "memory");        // no LDS reads hoisted above the wait

    const float* cur = &sbuf[c & 1][0];
    #pragma unroll 8
    for (int ts = 0; ts < CHUNK; ++ts) {
      const int t = c * CHUNK + ts + 1;   // output timestep for this row
      const f32x2 zin = *(const f32x2*)&cur[ts * HT + VEC * tid];
      const float r0 = (V.x > 1.0f) ? 1.0f : 0.0f;
      const float r1 = (V.y > 1.0f) ? 1.0f : 0.0f;
      V.x = __builtin_fmaf(0.9f, V.x, zin.x) - r0;  // beta*V + z - reset*thr
      V.y = __builtin_fmaf(0.9f, V.y, zin.y) - r1;
      f32x2 spk;
      spk.x = (V.x > 1.0f) ? 1.0f : 0.0f;           // heaviside(V - thr)
      spk.y = (V.y > 1.0f) ? 1.0f : 0.0f;
      if (t < kT)
        __builtin_nontemporal_store(
            spk, (f32x2*)&obase[(size_t)t * kH + VEC * tid]);
    }
  }
}

extern "C" void kernel_launch(void* const* d_in, const int* in_sizes, int n_in,
                              void* d_out, int out_size, void* d_ws, size_t ws_size,
                              hipStream_t stream) {
  (void)in_sizes; (void)n_in; (void)out_size; (void)d_ws; (void)ws_size;
  const float* z = (const float*)d_in[0];
  float* out = (float*)d_out;
  dim3 grid(kB, kH / HT);                 // 32 x 8 = 256 single-wave workgroups
  lif_tdm_kernel<<<grid, NT, 0, stream>>>(z, out);
}